// SimpleLSTM_58531814309925
// MI455X (gfx1250) — compile-verified
//
#include <hip/hip_runtime.h>

// ---------------------------------------------------------------------------
// SimpleLSTM, H=1: one thread per batch row, serial scan over T=2048.
// Chain-latency-optimized for gfx1250:
//   - every activation is one native v_tanh_f32 (sigmoid via prescaled
//     weights: sigmoid(z) = 0.5*tanh(0.5*z)+0.5, with the 0.5 folded into
//     the SGPR weights/biases so the chain is fma -> tanh with no mul)
//   - float4 x loads give 4 steps of off-chain pre-gate ILP
//   - global_prefetch_b8 one 128B line ahead
//   - non-temporal B128 output stores (write-once stream)
// ---------------------------------------------------------------------------

typedef float v4f __attribute__((ext_vector_type(4)));

#ifndef __has_builtin
#define __has_builtin(x) 0
#endif

__device__ __forceinline__ float tanh_native(float x) {
#if __has_builtin(__builtin_amdgcn_tanhf)
    return __builtin_amdgcn_tanhf(x);   // gfx1250 V_TANH_F32
#else
    float r;
    asm("v_tanh_f32 %0, %1" : "=v"(r) : "v"(x));
    return r;
#endif
}

// One LSTM timestep.
//   pre_i/f/o are PRESCALED by 0.5:  pre = 0.5*(x*w_ih + b)
//   wh_i/f/o are PRESCALED by 0.5
//   pre_g / wh_g are unscaled.
// Critical path per step: fma -> tanh -> fma -> fma -> tanh -> mul.
__device__ __forceinline__ float lstm_step(float pre_i, float pre_f,
                                           float pre_g, float pre_o,
                                           float wh_i, float wh_f,
                                           float wh_g, float wh_o,
                                           float& h, float& c) {
    float ti = tanh_native(__builtin_fmaf(h, wh_i, pre_i));  // tanh(0.5*zi)
    float tf = tanh_native(__builtin_fmaf(h, wh_f, pre_f));
    float gg = tanh_native(__builtin_fmaf(h, wh_g, pre_g));  // tanh(zg)
    float to = tanh_native(__builtin_fmaf(h, wh_o, pre_o));
    float gi = __builtin_fmaf(0.5f, ti, 0.5f);               // sigmoid(zi)
    float gf = __builtin_fmaf(0.5f, tf, 0.5f);
    float go = __builtin_fmaf(0.5f, to, 0.5f);               // off c-chain
    c = __builtin_fmaf(gf, c, gi * gg);
    h = go * tanh_native(c);
    return h;
}

__global__ __launch_bounds__(128) void lstm_h1_kernel(
    const float* __restrict__ x,     // [B, T] (H=1 squeezed)
    const float* __restrict__ w_ih,  // [4]  gate order: i, f, g, o
    const float* __restrict__ w_hh,  // [4]
    const float* __restrict__ b_ih,  // [4]
    const float* __restrict__ b_hh,  // [4]
    float* __restrict__ out,         // [B, T]
    float* __restrict__ h_n,         // [B]
    float* __restrict__ c_n,         // [B]
    int B, int T) {
    const int b = blockIdx.x * blockDim.x + threadIdx.x;
    if (b >= B) return;

    // Uniform weights -> SGPRs. Fold the sigmoid 0.5 prescale into i/f/o.
    const float wi_i = 0.5f * w_ih[0], wi_f = 0.5f * w_ih[1];
    const float wi_g =        w_ih[2], wi_o = 0.5f * w_ih[3];
    const float wh_i = 0.5f * w_hh[0], wh_f = 0.5f * w_hh[1];
    const float wh_g =        w_hh[2], wh_o = 0.5f * w_hh[3];
    const float bi = 0.5f * (b_ih[0] + b_hh[0]);
    const float bf = 0.5f * (b_ih[1] + b_hh[1]);
    const float bg =        (b_ih[2] + b_hh[2]);
    const float bo = 0.5f * (b_ih[3] + b_hh[3]);

    const v4f* __restrict__ xrow = (const v4f*)(x + (size_t)b * (size_t)T);
    v4f* __restrict__ orow       = (v4f*)(out + (size_t)b * (size_t)T);

    float h = 0.0f, c = 0.0f;
    const int n4 = T >> 2;  // T = 2048, divisible by 4

    for (int q = 0; q < n4; ++q) {
        v4f xv = xrow[q];
        if (q + 8 < n4) {
            // gfx1250 global_prefetch_b8: pull the next 128B line of x
            __builtin_prefetch((const void*)&xrow[q + 8], 0, 3);
        }

        // h-independent pre-gates for 4 steps (fills trans-latency stalls)
        v4f pre_i = xv * wi_i + bi;
        v4f pre_f = xv * wi_f + bf;
        v4f pre_g = xv * wi_g + bg;
        v4f pre_o = xv * wi_o + bo;

        v4f hv;
        hv.x = lstm_step(pre_i.x, pre_f.x, pre_g.x, pre_o.x,
                         wh_i, wh_f, wh_g, wh_o, h, c);
        hv.y = lstm_step(pre_i.y, pre_f.y, pre_g.y, pre_o.y,
                         wh_i, wh_f, wh_g, wh_o, h, c);
        hv.z = lstm_step(pre_i.z, pre_f.z, pre_g.z, pre_o.z,
                         wh_i, wh_f, wh_g, wh_o, h, c);
        hv.w = lstm_step(pre_i.w, pre_f.w, pre_g.w, pre_o.w,
                         wh_i, wh_f, wh_g, wh_o, h, c);

        // Write-once stream: non-temporal B128 store, keep caches for x
        __builtin_nontemporal_store(hv, &orow[q]);
    }

    h_n[b] = h;
    c_n[b] = c;
}

extern "C" void kernel_launch(void* const* d_in, const int* in_sizes, int n_in,
                              void* d_out, int out_size, void* d_ws, size_t ws_size,
                              hipStream_t stream) {
    const float* x    = (const float*)d_in[0];  // [B, T, 1]
    const float* w_ih = (const float*)d_in[1];  // [4,1]
    const float* w_hh = (const float*)d_in[2];  // [4,1]
    const float* b_ih = (const float*)d_in[3];  // [4]
    const float* b_hh = (const float*)d_in[4];  // [4]

    const int T = 2048;
    const int B = in_sizes[0] / T;

    float* out = (float*)d_out;                      // [B, T]
    float* h_n = out + (size_t)B * (size_t)T;        // [1, B, 1]
    float* c_n = h_n + B;                            // [1, B, 1]

    dim3 block(128);
    dim3 grid((B + 127) / 128);
    lstm_h1_kernel<<<grid, block, 0, stream>>>(x, w_ih, w_hh, b_ih, b_hh,
                                               out, h_n, c_n, B, T);
}